// GNNet_26474178412658
// MI455X (gfx1250) — compile-verified
//
#include <hip/hip_runtime.h>

// GraphSAGE 3-layer forward for MI455X (gfx1250, wave32).
// deg once -> per layer: [memset agg, edge scatter f32-atomics, fused WMMA GEMM
// (self + neigh, deg-scale + bias + relu in epilogue)] -> skinny output layer.

typedef float v2f __attribute__((ext_vector_type(2)));
typedef float v8f __attribute__((ext_vector_type(8)));

#define DIN 128

// ---------------- degree ----------------
__global__ void __launch_bounds__(256)
deg_count_kernel(const int* __restrict__ dst, float* __restrict__ deg, int nE) {
    int e = blockIdx.x * blockDim.x + threadIdx.x;
    if (e < nE) atomicAdd(&deg[dst[e]], 1.0f);
}

__global__ void __launch_bounds__(256)
deg_inv_kernel(float* __restrict__ deg, int n) {
    int i = blockIdx.x * blockDim.x + threadIdx.x;
    if (i < n) deg[i] = 1.0f / fmaxf(deg[i], 1.0f);
}

// ---------------- edge scatter: one wave32 per edge ----------------
// 32 lanes x float4 = 128 floats: coalesced 512B gather of h[src],
// fire-and-forget global_atomic_add_f32 into agg[dst] (L2-resident).
__global__ void __launch_bounds__(256)
scatter_kernel(const float* __restrict__ h, const int* __restrict__ src,
               const int* __restrict__ dst, float* __restrict__ agg, int nE) {
    int wave = blockIdx.x * (blockDim.x >> 5) + (threadIdx.x >> 5);
    int lane = threadIdx.x & 31;
    if (wave >= nE) return;                       // wave-uniform exit
    int s = src[wave];
    int d = dst[wave];
    float4 v = reinterpret_cast<const float4*>(h + (size_t)s * DIN)[lane];
    float* ap = agg + (size_t)d * DIN + lane * 4;
    atomicAdd(ap + 0, v.x);
    atomicAdd(ap + 1, v.y);
    atomicAdd(ap + 2, v.z);
    atomicAdd(ap + 3, v.w);
}

// ---------------- fused SAGE GEMM via V_WMMA_F32_16X16X4_F32 ----------------
// out[m][n] = relu?( h[m]@Ws + deginv[m]*(agg[m]@Wn) + b[n] )
// Wave strip: 16 rows x 64 cols (4 tiles x 2 accumulators). M assumed %16==0.
__global__ void __launch_bounds__(256)
sage_gemm_kernel(const float* __restrict__ h, const float* __restrict__ agg,
                 const float* __restrict__ deginv,
                 const float* __restrict__ Ws, const float* __restrict__ Wn,
                 const float* __restrict__ bias, float* __restrict__ out,
                 int M, int do_relu) {
    int wave = blockIdx.x * (blockDim.x >> 5) + (threadIdx.x >> 5);
    int lane = threadIdx.x & 31;
    int m0 = (wave >> 1) * 16;
    int n0 = (wave & 1) * 64;
    if (m0 >= M) return;                          // wave-uniform: EXEC stays all-1s

    int lo = lane & 15;
    int hi = lane >> 4;                           // half-wave select
    const float* hrow = h   + (size_t)(m0 + lo) * DIN;
    const float* arow = agg + (size_t)(m0 + lo) * DIN;

    v8f accS[4] = {};
    v8f accN[4] = {};

    // K loop: 32 steps of K=4, 8 WMMAs per step (4 col tiles x {self,neigh})
    #pragma unroll 4
    for (int k = 0; k < DIN; k += 4) {
        int kk = k + 2 * hi;                      // ISA A-layout: v0/v1 = K{kk,kk+1}
        v2f aH = *reinterpret_cast<const v2f*>(hrow + kk);
        v2f aA = *reinterpret_cast<const v2f*>(arow + kk);
        #pragma unroll
        for (int t = 0; t < 4; ++t) {
            int col = n0 + t * 16 + lo;
            v2f bS, bN;                           // ISA B-layout: rows kk, kk+1
            bS.x = Ws[(size_t)kk * DIN + col];
            bS.y = Ws[(size_t)(kk + 1) * DIN + col];
            bN.x = Wn[(size_t)kk * DIN + col];
            bN.y = Wn[(size_t)(kk + 1) * DIN + col];
            accS[t] = __builtin_amdgcn_wmma_f32_16x16x4_f32(
                false, aH, false, bS, (short)0, accS[t], false, false);
            accN[t] = __builtin_amdgcn_wmma_f32_16x16x4_f32(
                false, aA, false, bN, (short)0, accN[t], false, false);
        }
    }

    // Row-scale (mean) commutes with GEMM: apply deginv in epilogue.
    float dv[8];
    #pragma unroll
    for (int j = 0; j < 8; ++j) dv[j] = deginv[m0 + j + 8 * hi];

    #pragma unroll
    for (int t = 0; t < 4; ++t) {
        int col = n0 + t * 16 + lo;
        float bn = bias[col];
        #pragma unroll
        for (int j = 0; j < 8; ++j) {             // C layout: VGPR j -> M = j + 8*hi
            int mr = m0 + j + 8 * hi;
            float val = accS[t][j] + dv[j] * accN[t][j] + bn;
            if (do_relu) val = fmaxf(val, 0.0f);
            out[(size_t)mr * DIN + col] = val;
        }
    }
}

// ---------------- final layer (128 -> 2): one wave per node ----------------
__global__ void __launch_bounds__(256)
out_layer_kernel(const float* __restrict__ h, const float* __restrict__ agg,
                 const float* __restrict__ deginv,
                 const float* __restrict__ Ws, const float* __restrict__ Wn,
                 const float* __restrict__ bias, float* __restrict__ out, int M) {
    int w = blockIdx.x * (blockDim.x >> 5) + (threadIdx.x >> 5);
    int lane = threadIdx.x & 31;
    if (w >= M) return;
    float4 hv = reinterpret_cast<const float4*>(h   + (size_t)w * DIN)[lane];
    float4 av = reinterpret_cast<const float4*>(agg + (size_t)w * DIN)[lane];
    float di = deginv[w];
    int k = lane * 4;
    float a0 = hv.x * Ws[(k + 0) * 2 + 0] + hv.y * Ws[(k + 1) * 2 + 0] +
               hv.z * Ws[(k + 2) * 2 + 0] + hv.w * Ws[(k + 3) * 2 + 0];
    float a1 = hv.x * Ws[(k + 0) * 2 + 1] + hv.y * Ws[(k + 1) * 2 + 1] +
               hv.z * Ws[(k + 2) * 2 + 1] + hv.w * Ws[(k + 3) * 2 + 1];
    a0 += di * (av.x * Wn[(k + 0) * 2 + 0] + av.y * Wn[(k + 1) * 2 + 0] +
                av.z * Wn[(k + 2) * 2 + 0] + av.w * Wn[(k + 3) * 2 + 0]);
    a1 += di * (av.x * Wn[(k + 0) * 2 + 1] + av.y * Wn[(k + 1) * 2 + 1] +
                av.z * Wn[(k + 2) * 2 + 1] + av.w * Wn[(k + 3) * 2 + 1]);
    #pragma unroll
    for (int off = 16; off > 0; off >>= 1) {
        a0 += __shfl_xor(a0, off, 32);
        a1 += __shfl_xor(a1, off, 32);
    }
    if (lane == 0) {
        out[(size_t)w * 2 + 0] = a0 + bias[0];
        out[(size_t)w * 2 + 1] = a1 + bias[1];
    }
}

extern "C" void kernel_launch(void* const* d_in, const int* in_sizes, int n_in,
                              void* d_out, int out_size, void* d_ws, size_t ws_size,
                              hipStream_t stream) {
    const float* x   = (const float*)d_in[0];
    const int*   src = (const int*)d_in[1];
    const int*   dst = (const int*)d_in[2];
    const float* Ws0 = (const float*)d_in[3];
    const float* Wn0 = (const float*)d_in[4];
    const float* b0  = (const float*)d_in[5];
    const float* Ws1 = (const float*)d_in[6];
    const float* Wn1 = (const float*)d_in[7];
    const float* b1  = (const float*)d_in[8];
    const float* Ws2 = (const float*)d_in[9];
    const float* Wn2 = (const float*)d_in[10];
    const float* b2  = (const float*)d_in[11];

    int M  = in_sizes[0] / DIN;   // 50000 nodes
    int nE = in_sizes[1];         // 600000 edges

    char* ws = (char*)d_ws;
    float* deg = (float*)ws; ws += (size_t)M * sizeof(float);
    float* agg = (float*)ws; ws += (size_t)M * DIN * sizeof(float);
    float* hA  = (float*)ws; ws += (size_t)M * DIN * sizeof(float);
    float* hB  = (float*)ws;

    const size_t featBytes = (size_t)M * DIN * sizeof(float);
    const int scatterBlocks = (nE + 7) / 8;                 // 1 wave32 / edge
    const int gemmWaves = ((M + 15) / 16) * 2;              // 16 rows x 64 cols / wave
    const int gemmBlocks = (gemmWaves + 7) / 8;
    const int nodeWaveBlocks = (M + 7) / 8;

    // degree (structure-only, once)
    hipMemsetAsync(deg, 0, (size_t)M * sizeof(float), stream);
    deg_count_kernel<<<(nE + 255) / 256, 256, 0, stream>>>(dst, deg, nE);
    deg_inv_kernel<<<(M + 255) / 256, 256, 0, stream>>>(deg, M);

    // layer 0: x -> hA (relu fused)
    hipMemsetAsync(agg, 0, featBytes, stream);
    scatter_kernel<<<scatterBlocks, 256, 0, stream>>>(x, src, dst, agg, nE);
    sage_gemm_kernel<<<gemmBlocks, 256, 0, stream>>>(x, agg, deg, Ws0, Wn0, b0, hA, M, 1);

    // layer 1: hA -> hB (relu fused)
    hipMemsetAsync(agg, 0, featBytes, stream);
    scatter_kernel<<<scatterBlocks, 256, 0, stream>>>(hA, src, dst, agg, nE);
    sage_gemm_kernel<<<gemmBlocks, 256, 0, stream>>>(hA, agg, deg, Ws1, Wn1, b1, hB, M, 1);

    // layer 2: hB -> out (N x 2)
    hipMemsetAsync(agg, 0, featBytes, stream);
    scatter_kernel<<<scatterBlocks, 256, 0, stream>>>(hB, src, dst, agg, nE);
    out_layer_kernel<<<nodeWaveBlocks, 256, 0, stream>>>(hB, agg, deg, Ws2, Wn2, b2,
                                                         (float*)d_out, M);
}